// GNNModel_14328010899631
// MI455X (gfx1250) — compile-verified
//
#include <hip/hip_runtime.h>

// ---------------------------------------------------------------------------
// 3-layer GraphConv GNN for MI455X (gfx1250, wave32).
//
//   out_i = W_root^T x_i + W_rel^T (sum_{j->i} x_j) + b        (PyG GraphConv)
//
// MI455X reasoning:
//  * 192 MB L2: all node-feature buffers (x 25.6MB, agg 51.2MB, h 51.2MB)
//    are L2-resident -> scatter-add runs at L2 atomic throughput; only the
//    edge list streams from HBM (~38 MB total @ 23.3 TB/s, negligible).
//  * GEMMs (~12.8 GFLOP total) are memory-bound; use V_WMMA_F32_16X16X4_F32
//    so the matrix pipe carries them in exact f32 reference precision.
//    M=100000 = 6250*16, N in {64,128}, K in {64,128}: no tail handling.
//  * Scatter-add uses native global_atomic_add_f32 (unsafeAtomicAdd), float4
//    gathers, one 16/32-lane group per edge.
// ---------------------------------------------------------------------------

typedef __attribute__((ext_vector_type(2))) float v2f;
typedef __attribute__((ext_vector_type(8))) float v8f;

#define N_NODES 100000
#define N_EDGES 1600000

// ---------------------------------------------------------------- zero fill
__global__ void zero_f4_kernel(float4* __restrict__ p, long long n4) {
  long long i = (long long)blockIdx.x * blockDim.x + threadIdx.x;
  long long stride = (long long)gridDim.x * blockDim.x;
  const float4 z = make_float4(0.f, 0.f, 0.f, 0.f);
  for (; i < n4; i += stride) p[i] = z;
}

// ------------------------------------------------------------- scatter-add
// agg[dst[e]] += x[src[e]]  for all edges, feature dim D (64 or 128).
// D/4 lanes per edge, one float4 per lane. Atomics resolve in L2 (RT hint).
template <int D>
__global__ void scatter_add_kernel(const float* __restrict__ x,
                                   const int* __restrict__ src,
                                   const int* __restrict__ dst,
                                   float* __restrict__ agg) {
  constexpr int CPE = D / 4;  // float4 chunks per edge (power of two)
  const long long total = (long long)N_EDGES * CPE;
  long long i = (long long)blockIdx.x * blockDim.x + threadIdx.x;
  const long long stride = (long long)gridDim.x * blockDim.x;
  for (; i < total; i += stride) {
    const int e = (int)(i / CPE);
    const int c = (int)(i % CPE);
    const int s = src[e];
    const int t = dst[e];
    const float4 v = *(const float4*)(x + (size_t)s * D + 4 * c);
    float* p = agg + (size_t)t * D + 4 * c;
    unsafeAtomicAdd(p + 0, v.x);   // -> global_atomic_add_f32
    unsafeAtomicAdd(p + 1, v.y);
    unsafeAtomicAdd(p + 2, v.z);
    unsafeAtomicAdd(p + 3, v.w);
  }
}

// ------------------------------------------------------- fused dual WMMA GEMM
// out = A1 @ B1 + A2 @ B2 + bias  (optionally ReLU), all f32.
// A*: [100000 x K] row-major, B*: [K x Ncols] row-major, out: [100000 x Ncols].
// One wave computes one 16x16 output tile via V_WMMA_F32_16X16X4_F32.
//
// Per-lane operand layout (ISA 7.12.2, 32-bit, wave32):
//   A (16x4):  lane l -> row m = l&15, half h = l>>4; VGPR v holds A[m][2h+v]
//   B (4x16):  lane l -> col n = l&15, half h;        VGPR v holds B[2h+v][n]
//   C/D(16x16):lane l -> col n = l&15, half h;        VGPR v holds D[v+8h][n]
__global__ void gemm_dual_wmma_kernel(const float* __restrict__ A1,
                                      const float* __restrict__ B1,
                                      const float* __restrict__ A2,
                                      const float* __restrict__ B2,
                                      const float* __restrict__ bias,
                                      float* __restrict__ out,
                                      int Ncols, int K, int relu) {
  const int lane = threadIdx.x & 31;
  const int wave = (blockIdx.x * blockDim.x + threadIdx.x) >> 5;
  const int ntn = Ncols >> 4;               // tiles along N
  const int tiles = (N_NODES >> 4) * ntn;   // 6250 * ntn
  if (wave >= tiles) return;                // uniform per wave: EXEC stays all-1s
  const int tm = wave / ntn;
  const int tn = wave - tm * ntn;

  const int lm = lane & 15;   // A-row / B-col / C-col within tile
  const int lh = lane >> 4;   // lane half: K-pair select (A/B), M-half (C/D)
  const int row0 = tm << 4;
  const int col = (tn << 4) + lm;

  v8f acc = {0.f, 0.f, 0.f, 0.f, 0.f, 0.f, 0.f, 0.f};

  const float* a1row = A1 + (size_t)(row0 + lm) * K + 2 * lh;   // 8B aligned
  const float* a2row = A2 + (size_t)(row0 + lm) * K + 2 * lh;
  const float* b1col = B1 + (size_t)(2 * lh) * Ncols + col;
  const float* b2col = B2 + (size_t)(2 * lh) * Ncols + col;

  // Pass 1: A1 @ B1  (aggregated messages x W_rel)
  for (int k = 0; k < K; k += 4) {
    v2f a = *(const v2f*)(a1row + k);                 // A[m][k+2h], A[m][k+2h+1]
    v2f b;
    b.x = b1col[(size_t)k * Ncols];                   // B[k+2h  ][n]
    b.y = b1col[(size_t)(k + 1) * Ncols];             // B[k+2h+1][n]
    acc = __builtin_amdgcn_wmma_f32_16x16x4_f32(
        /*neg_a=*/false, a, /*neg_b=*/false, b,
        /*c_mod=*/(short)0, acc, /*reuse_a=*/false, /*reuse_b=*/false);
  }
  // Pass 2: A2 @ B2  (node features x W_root), same K
  for (int k = 0; k < K; k += 4) {
    v2f a = *(const v2f*)(a2row + k);
    v2f b;
    b.x = b2col[(size_t)k * Ncols];
    b.y = b2col[(size_t)(k + 1) * Ncols];
    acc = __builtin_amdgcn_wmma_f32_16x16x4_f32(
        false, a, false, b, (short)0, acc, false, false);
  }

  const float bv = bias[col];
#pragma unroll
  for (int v = 0; v < 8; ++v) {
    float val = acc[v] + bv;
    if (relu) val = fmaxf(val, 0.f);
    out[(size_t)(row0 + v + 8 * lh) * Ncols + col] = val;  // row-coalesced
  }
}

// ---------------------------------------------------------------------------
extern "C" void kernel_launch(void* const* d_in, const int* in_sizes, int n_in,
                              void* d_out, int out_size, void* d_ws, size_t ws_size,
                              hipStream_t stream) {
  // setup_inputs() order:
  const float* x     = (const float*)d_in[0];   // [100000 x 64]
  const int*   ei    = (const int*)d_in[1];     // [2 x 1600000] (int32 per harness mapping)
  const float* W1rel = (const float*)d_in[2];   // [64 x 128]
  const float* b1    = (const float*)d_in[3];   // [128]
  const float* W1rt  = (const float*)d_in[4];   // [64 x 128]
  const float* W2rel = (const float*)d_in[5];   // [128 x 128]
  const float* b2    = (const float*)d_in[6];   // [128]
  const float* W2rt  = (const float*)d_in[7];   // [128 x 128]
  const float* W3rel = (const float*)d_in[8];   // [128 x 64]
  const float* b3    = (const float*)d_in[9];   // [64]
  const float* W3rt  = (const float*)d_in[10];  // [128 x 64]
  float* out = (float*)d_out;                   // [100000 x 64]

  const int* src = ei;            // edge_index[0]
  const int* dst = ei + N_EDGES;  // edge_index[1]

  // Workspace: agg | h1 | h2, each 100000 x 128 f32 (51.2 MB -> 153.6 MB total).
  float* agg = (float*)d_ws;
  float* h1  = agg + (size_t)N_NODES * 128;
  float* h2  = h1 + (size_t)N_NODES * 128;

  const int BLK = 256;
  const int WPB = BLK / 32;  // waves per block

  // ---------------- Layer 1: 64 -> 128, ReLU
  {
    long long n4 = (long long)N_NODES * 64 / 4;
    zero_f4_kernel<<<(int)((n4 + BLK - 1) / BLK), BLK, 0, stream>>>((float4*)agg, n4);
    long long work = (long long)N_EDGES * (64 / 4);
    scatter_add_kernel<64><<<(int)((work + BLK - 1) / BLK), BLK, 0, stream>>>(x, src, dst, agg);
    int tiles = (N_NODES / 16) * (128 / 16);
    gemm_dual_wmma_kernel<<<(tiles + WPB - 1) / WPB, BLK, 0, stream>>>(
        agg, W1rel, x, W1rt, b1, h1, /*Ncols=*/128, /*K=*/64, /*relu=*/1);
  }

  // ---------------- Layer 2: 128 -> 128, ReLU
  {
    long long n4 = (long long)N_NODES * 128 / 4;
    zero_f4_kernel<<<(int)((n4 + BLK - 1) / BLK), BLK, 0, stream>>>((float4*)agg, n4);
    long long work = (long long)N_EDGES * (128 / 4);
    scatter_add_kernel<128><<<(int)((work + BLK - 1) / BLK), BLK, 0, stream>>>(h1, src, dst, agg);
    int tiles = (N_NODES / 16) * (128 / 16);
    gemm_dual_wmma_kernel<<<(tiles + WPB - 1) / WPB, BLK, 0, stream>>>(
        agg, W2rel, h1, W2rt, b2, h2, /*Ncols=*/128, /*K=*/128, /*relu=*/1);
  }

  // ---------------- Layer 3: 128 -> 64, no activation
  {
    long long n4 = (long long)N_NODES * 128 / 4;
    zero_f4_kernel<<<(int)((n4 + BLK - 1) / BLK), BLK, 0, stream>>>((float4*)agg, n4);
    long long work = (long long)N_EDGES * (128 / 4);
    scatter_add_kernel<128><<<(int)((work + BLK - 1) / BLK), BLK, 0, stream>>>(h2, src, dst, agg);
    int tiles = (N_NODES / 16) * (64 / 16);
    gemm_dual_wmma_kernel<<<(tiles + WPB - 1) / WPB, BLK, 0, stream>>>(
        agg, W3rel, h2, W3rt, b3, out, /*Ncols=*/64, /*K=*/128, /*relu=*/0);
  }
}